// SpanRepresentation_35553739276881
// MI455X (gfx1250) — compile-verified
//
#include <hip/hip_runtime.h>
#include <stdint.h>
#include <stddef.h>

// ---------------------------------------------------------------------------
// SpanRepresentation: out[b,n] = concat(x[b,start(n)], x[b,end(n)], emb[bucket(n)])
// L=512, widths 1..8 -> N=4068 spans, H=768, W=25, row=1561 f32.
// Pure gather/concat: ~406 MB of write-once stores; x (25MB) is L2-resident
// (re-read ~16x). Store-bandwidth bound (~18us floor @ 23.3 TB/s).
// gfx1250 paths: global_load_async_to_lds_b32 (ASYNCcnt DMA) stages the
// width-embedding table in LDS; s_wait_asynccnt fence; b128 RT loads for x;
// non-temporal b128 stores for the streamed output (keep L2 for x).
// ---------------------------------------------------------------------------

#define SEQ_L     512
#define HID       768
#define W_EMB     25
#define N_BUCKET  14
#define N_SPANS   4068                 // sum_{w=1..8} (513-w)
#define ROW       (2 * HID + W_EMB)    // 1561
#define SPB       8                    // spans per block
#define TPB       256

typedef float f4 __attribute__((ext_vector_type(4)));
typedef f4 f4_a4 __attribute__((aligned(4)));   // 4B-aligned 128-bit access

__device__ __forceinline__ void decode_span(int n, int& w, int& st) {
    // width w in 1..8, count at width w is (SEQ_L - w + 1)
    int rem = n, cnt = SEQ_L;
    w = 1;
    while (rem >= cnt) { rem -= cnt; --cnt; ++w; }
    st = rem;
}

__global__ __launch_bounds__(TPB) void span_rep_kernel(
    const float* __restrict__ x,      // [B, 512, 768]
    const float* __restrict__ emb,    // [14, 25]
    float* __restrict__ out)          // [B, 4068, 1561]
{
    __shared__ float semb[N_BUCKET * W_EMB];   // 350 floats = 1400 B
    const int tid = threadIdx.x;

    // --- Stage width-embedding table into LDS via gfx1250 async DMA ---------
    {
        // Flat shared pointer: low 32 bits == LDS byte offset (ISA 10.2).
        uint32_t lds_base = (uint32_t)(uintptr_t)(void*)&semb[0];
        for (int i = tid; i < N_BUCKET * W_EMB; i += TPB) {
            uint32_t lds_addr = lds_base + (uint32_t)(i * 4);
            uint64_t gaddr    = (uint64_t)(uintptr_t)(emb + i);
            asm volatile("global_load_async_to_lds_b32 %0, %1, off"
                         :: "v"(lds_addr), "v"(gaddr) : "memory");
        }
        asm volatile("s_wait_asynccnt 0" ::: "memory");
        __syncthreads();
    }

    const int b  = blockIdx.y;
    const int n0 = blockIdx.x * SPB;

    for (int s = 0; s < SPB; ++s) {
        const int n = n0 + s;
        if (n >= N_SPANS) break;

        int w, st;
        decode_span(n, w, st);             // wave-uniform -> SALU
        const int en     = st + w - 1;
        const int bucket = (w <= 5) ? w : (w - 1);

        const float* __restrict__ s0 = x + ((size_t)b * SEQ_L + st) * HID; // 16B aligned
        const float* __restrict__ s1 = x + ((size_t)b * SEQ_L + en) * HID; // 16B aligned
        float* __restrict__ d = out + ((size_t)b * N_SPANS + n) * (size_t)ROW; // 4B aligned

        // 2*768 floats = 384 float4 chunks; RT b128 loads (L2-resident x),
        // non-temporal b128 stores (write-once stream, keep L2 for x).
        #pragma unroll 2
        for (int i = tid; i < 2 * (HID / 4); i += TPB) {
            f4 v;
            if (i < HID / 4) v = *(const f4*)(s0 + 4 * i);
            else             v = *(const f4*)(s1 + 4 * i - HID);
            __builtin_nontemporal_store(v, (f4_a4*)(d + 4 * i));
        }
        // 25-float width-embedding tail from LDS.
        if (tid < W_EMB)
            __builtin_nontemporal_store(semb[bucket * W_EMB + tid], d + 2 * HID + tid);
    }
}

__global__ __launch_bounds__(TPB) void span_indices_kernel(float* __restrict__ idx_out)
{
    const int n = blockIdx.x * blockDim.x + threadIdx.x;
    if (n >= N_SPANS) return;
    int w, st;
    decode_span(n, w, st);
    __builtin_nontemporal_store((float)st,          idx_out + 2 * n + 0); // exact: <= 511
    __builtin_nontemporal_store((float)(st + w - 1), idx_out + 2 * n + 1);
}

extern "C" void kernel_launch(void* const* d_in, const int* in_sizes, int n_in,
                              void* d_out, int out_size, void* d_ws, size_t ws_size,
                              hipStream_t stream)
{
    (void)d_ws; (void)ws_size; (void)out_size; (void)n_in;

    const float* x   = (const float*)d_in[0];   // [B,512,768] f32
    const float* emb = (const float*)d_in[1];   // [14,25] f32
    // d_in[2] = batch_max_seq_len (==512, static in the reference)

    const int B = in_sizes[0] / (SEQ_L * HID);  // 16

    float* out     = (float*)d_out;
    float* idx_out = out + (size_t)B * N_SPANS * (size_t)ROW;

    dim3 grid((N_SPANS + SPB - 1) / SPB, B, 1);
    span_rep_kernel<<<grid, TPB, 0, stream>>>(x, emb, out);

    span_indices_kernel<<<(N_SPANS + TPB - 1) / TPB, TPB, 0, stream>>>(idx_out);
}